// OrographyLatent_51737176048339
// MI455X (gfx1250) — compile-verified
//
#include <hip/hip_runtime.h>
#include <hip/hip_bf16.h>

// Problem constants (match reference)
#define NPIX   65536   // H*W = 256*256
#define KNB    25      // (2*RADIUS+1)^2
#define MAPD   64      // MAP_DIM
#define NB     8       // batch B
#define KSTEPS 7       // ceil(25/4) K-steps of the 16x16x4 f32 WMMA
#define PIXPW  4       // pixels handled per wave
#define WAVES  8       // waves per 256-thread block

typedef __attribute__((ext_vector_type(2))) float v2f;
typedef __attribute__((ext_vector_type(8))) float v8f;

// One wave32 computes one pixel's 8x25x64 GEMM with V_WMMA_F32_16X16X4_F32:
//   D(16x16) = A(16x4) * B(4x16) + C, 4 N-tiles x 7 K-steps = 28 WMMAs/pixel.
__global__ __launch_bounds__(256)
void OrographyLatent_wmma_f32(const float* __restrict__ orog,      // (8, 65536)
                              const int*   __restrict__ nidx,      // (65536, 25)
                              const float* __restrict__ bias_low,  // (65536)
                              const float* __restrict__ bias_high, // (65536, 64)
                              const float* __restrict__ wmap,      // (65536, 25, 64)
                              float*       __restrict__ out)       // (8, 65536, 64)
{
    const int lane  = threadIdx.x & 31;
    const int wave  = threadIdx.x >> 5;
    const int wid   = blockIdx.x * WAVES + wave;
    const int m     = lane & 15;   // matrix row (batch; rows 8..15 are zero pad)
    const int khalf = lane >> 4;   // selects K pair {2*khalf, 2*khalf+1} per ISA layout

    for (int it = 0; it < PIXPW; ++it) {
        const int p = wid * PIXPW + it;
        const float* __restrict__ wp = wmap + (size_t)p * (KNB * MAPD);

        // Prefetch next pixel's 6.4 KB weight block (lowers to global_prefetch_b8).
        if (it + 1 < PIXPW) {
            const char* nf = (const char*)(wp + KNB * MAPD);
            __builtin_prefetch(nf + lane * 200, 0, 1);
        }

        // ---- A operands: gather_y in WMMA 16x4 f32 layout -------------------
        // lane l: M = l&15, a.x holds K = 4*ks + 2*khalf, a.y holds K+1.
        v2f a[KSTEPS];
        #pragma unroll
        for (int ks = 0; ks < KSTEPS; ++ks) {
            const int k0 = ks * 4 + 2 * khalf;
            const int k1 = k0 + 1;
            float v0 = 0.0f, v1 = 0.0f;
            if (m < NB) {
                if (k0 < KNB) {
                    const int g = nidx[p * KNB + k0];
                    v0 = orog[(size_t)m * NPIX + g] - bias_low[g];
                }
                if (k1 < KNB) {
                    const int g = nidx[p * KNB + k1];
                    v1 = orog[(size_t)m * NPIX + g] - bias_low[g];
                }
            }
            a[ks].x = v0;
            a[ks].y = v1;
        }

        // ---- C init: bias_high broadcast over all M rows --------------------
        // C/D VGPR j, lane l -> (M = j + 8*(l>=16), N = l&15); bias depends on N only.
        v8f acc[4];
        #pragma unroll
        for (int nt = 0; nt < 4; ++nt) {
            const float bh = bias_high[(size_t)p * MAPD + nt * 16 + m];
            acc[nt] = (v8f){bh, bh, bh, bh, bh, bh, bh, bh};
        }

        // ---- K loop: 7 steps x 4 N-tiles of fp32 WMMA -----------------------
        #pragma unroll
        for (int ks = 0; ks < KSTEPS; ++ks) {
            const int r0 = ks * 4 + 2 * khalf;   // B rows this lane supplies
            const int r1 = r0 + 1;
            #pragma unroll
            for (int nt = 0; nt < 4; ++nt) {
                const int n = nt * 16 + m;       // B column for this lane
                v2f b;
                b.x = (r0 < KNB) ? wp[r0 * MAPD + n] : 0.0f;
                b.y = (r1 < KNB) ? wp[r1 * MAPD + n] : 0.0f;
                // 8 args: (neg_a, A, neg_b, B, c_mod, C, reuse_a, reuse_b)
                acc[nt] = __builtin_amdgcn_wmma_f32_16x16x4_f32(
                    false, a[ks], false, b, (short)0, acc[nt], false, false);
            }
        }

        // ---- Store real rows 0..7 (lanes 0..15 hold M = 0..7 in VGPR j) -----
        if (lane < 16) {
            #pragma unroll
            for (int nt = 0; nt < 4; ++nt) {
                #pragma unroll
                for (int j = 0; j < NB; ++j) {
                    out[((size_t)j * NPIX + p) * MAPD + nt * 16 + lane] = acc[nt][j];
                }
            }
        }
    }
}

extern "C" void kernel_launch(void* const* d_in, const int* in_sizes, int n_in,
                              void* d_out, int out_size, void* d_ws, size_t ws_size,
                              hipStream_t stream) {
    // setup_inputs() order: orog_in, neighbor_indices, bias_low, bias_high, weight_map
    const float* orog      = (const float*)d_in[0];
    const int*   nidx      = (const int*)  d_in[1];
    const float* bias_low  = (const float*)d_in[2];
    const float* bias_high = (const float*)d_in[3];
    const float* wmap      = (const float*)d_in[4];
    float*       out       = (float*)d_out;

    // 65536 pixels / (8 waves * 4 pixels per wave) = 2048 blocks of 256 threads.
    dim3 grid(NPIX / (WAVES * PIXPW));
    dim3 block(WAVES * 32);
    hipLaunchKernelGGL(OrographyLatent_wmma_f32, grid, block, 0, stream,
                       orog, nidx, bias_low, bias_high, wmap, out);
}